// GraphConv_57664230916666
// MI455X (gfx1250) — compile-verified
//
#include <hip/hip_runtime.h>

#define N_NODES 100000
#define N_EDGES 1600000
#define D 128
#define M_TILES (N_NODES / 16)   // 6250 exactly

typedef __attribute__((ext_vector_type(2))) float v2f;
typedef __attribute__((ext_vector_type(8))) float v8f;

// ---------------- Kernel 1: zero the aggregation scratch ----------------
__global__ void zero_agg_kernel(float4* __restrict__ agg, int n4) {
    int i = blockIdx.x * blockDim.x + threadIdx.x;
    if (i < n4) {
        float4 z; z.x = 0.f; z.y = 0.f; z.z = 0.f; z.w = 0.f;
        agg[i] = z;
    }
}

// ---------------- Kernel 2: edge gather + atomic scatter-add ----------------
// One wave (32 lanes) per edge: lane l handles channels 4l..4l+3.
// x (51.2MB) and agg (51.2MB) both fit in the 192MB L2, so the row gather
// mostly hits L2 and the f32 atomics resolve at the L2 atomic units.
__global__ __launch_bounds__(256) void scatter_kernel(
    const float* __restrict__ x,
    const int* __restrict__ src,
    const int* __restrict__ dst,
    float* __restrict__ agg)
{
    int wave = (blockIdx.x * blockDim.x + threadIdx.x) >> 5;
    int lane = threadIdx.x & 31;
    if (wave >= N_EDGES) return;

    int s = src[wave];
    int d = dst[wave];

    const float4 v = reinterpret_cast<const float4*>(x + (size_t)s * D)[lane];
    float* p = agg + (size_t)d * D + lane * 4;
    atomicAdd(p + 0, v.x);
    atomicAdd(p + 1, v.y);
    atomicAdd(p + 2, v.z);
    atomicAdd(p + 3, v.w);
}

// ---------------- Kernel 3: out = agg @ W^T + b via V_WMMA_F32_16X16X4_F32 ----
// One wave computes a 16-row x 128-col output slab: 8 C tiles (v8f each),
// K accumulated 4 at a time (32 WMMA steps x 8 N-tiles = 256 wmma per wave).
//
// LDS holds W pair-interleaved: Wp[kp][n][v] = W[n][2*kp + v]  (64 KB).
// A B-fragment (two consecutive K values for a fixed column n) is then one
// aligned ds_load_b64 straight into an even-aligned v2f -> no repack movs.
__global__ __launch_bounds__(256) void gemm_wmma_kernel(
    const float* __restrict__ agg,
    const float* __restrict__ W,
    const float* __restrict__ bias,
    float* __restrict__ out)
{
    __shared__ float Wp[D * D];  // 64 KB, indexed as (kp*128 + n)*2 + v

    // Cooperative pair-interleaved transpose load: coalesced global reads.
    for (int i = threadIdx.x; i < D * D; i += 256) {
        int n = i >> 7;      // output channel (row of W)
        int k = i & 127;     // input channel
        Wp[(k >> 1) * 256 + n * 2 + (k & 1)] = W[i];
    }
    __syncthreads();

    int lane   = threadIdx.x & 31;
    int waveId = blockIdx.x * 8 + (threadIdx.x >> 5);
    int rowBase = waveId * 16;
    if (rowBase >= N_NODES) return;   // wave-uniform; EXEC stays all-ones for WMMA

    int laneHi = lane >> 4;   // 0: lanes 0-15, 1: lanes 16-31
    int nlane  = lane & 15;
    int m      = nlane;       // A-matrix row held by this lane

    v8f zero = {0.f, 0.f, 0.f, 0.f, 0.f, 0.f, 0.f, 0.f};
    v8f c[8];
#pragma unroll
    for (int t = 0; t < 8; ++t) c[t] = zero;

    const float* arow = agg + (size_t)rowBase * D + (size_t)m * D;

    for (int kb = 0; kb < 32; ++kb) {
        // A fragment 16x4: lanes 0-15 hold K = 4kb+0,+1 ; lanes 16-31 hold K = 4kb+2,+3
        int k0  = kb * 4 + laneHi * 2;
        int kp0 = kb * 2 + laneHi;          // K-pair index for this lane
        v2f a;
        a.x = arow[k0];
        a.y = arow[k0 + 1];
#pragma unroll
        for (int t = 0; t < 8; ++t) {
            int n = t * 16 + nlane;
            // B fragment 4x16: VGPR v holds row K = k0 + v, col = nlane
            v2f b = *reinterpret_cast<const v2f*>(&Wp[kp0 * 256 + n * 2]);
            c[t] = __builtin_amdgcn_wmma_f32_16x16x4_f32(
                false, a, false, b, (short)0, c[t], false, false);
        }
    }

    // C/D 16x16 layout: VGPR r -> row = r + laneHi*8, col = nlane (within tile)
#pragma unroll
    for (int t = 0; t < 8; ++t) {
        int n = t * 16 + nlane;
        float bn = bias[n];
#pragma unroll
        for (int r = 0; r < 8; ++r) {
            int row = rowBase + r + laneHi * 8;
            out[(size_t)row * D + n] = c[t][r] + bn;
        }
    }
}

extern "C" void kernel_launch(void* const* d_in, const int* in_sizes, int n_in,
                              void* d_out, int out_size, void* d_ws, size_t ws_size,
                              hipStream_t stream) {
    const float* x   = (const float*)d_in[0];
    const int*   src = (const int*)d_in[1];
    const int*   dst = (const int*)d_in[2];
    const float* W   = (const float*)d_in[3];
    const float* b   = (const float*)d_in[4];
    float* out = (float*)d_out;
    float* agg = (float*)d_ws;   // N_NODES * D floats = 51.2 MB scratch

    // 1) agg = 0
    int n4 = N_NODES * D / 4;    // 3,200,000 float4
    zero_agg_kernel<<<(n4 + 255) / 256, 256, 0, stream>>>((float4*)agg, n4);

    // 2) scatter-add messages: one wave per edge -> 1.6M waves / 8 per block
    scatter_kernel<<<N_EDGES / 8, 256, 0, stream>>>(x, src, dst, agg);

    // 3) WMMA GEMM + bias: 6250 row-tiles, 8 waves per block
    gemm_wmma_kernel<<<(M_TILES + 7) / 8, 256, 0, stream>>>(agg, W, b, out);
}